// FALayer_91173565759775
// MI455X (gfx1250) — compile-verified
//
#include <hip/hip_runtime.h>
#include <math.h>

#define N_NODES 10000
#define N_EDGES 320000
#define D_FEAT  256

typedef __attribute__((ext_vector_type(2))) float v2f;
typedef __attribute__((ext_vector_type(8))) float v8f;

// ---------------------------------------------------------------------------
// Kernel 1: zero the per-dst histogram counters.
// ---------------------------------------------------------------------------
__global__ void k_zero(unsigned int* __restrict__ cnt) {
    int i = blockIdx.x * blockDim.x + threadIdx.x;
    if (i < N_NODES) cnt[i] = 0u;
}

// ---------------------------------------------------------------------------
// Kernel 2: node scores via V_WMMA_F32_16X16X4_F32.
// One wave computes score_dst/score_src for 16 nodes.
//   A[m][k] = h[node_m][kbase+k] * w[kbase+k]   (weight folded into A)
//   B       = all ones  -> D[m][n] = sum_k A[m][k]  (layout-robust reduction)
// A layout (ISA 7.12.2, 32-bit A 16x4): lanes 0-15 M=0..15 supply K=0,1;
// lanes 16-31 (same M) supply K=2,3. D layout: VGPR r = row M=r (+8 for
// lanes 16-31), N = lane%16.
// ---------------------------------------------------------------------------
__global__ void k_scores(const float* __restrict__ h,
                         const float* __restrict__ gw,   // [512]: w_dst | w_src
                         const float* __restrict__ gb,   // [1]
                         float* __restrict__ sd,
                         float* __restrict__ ss) {
    int wave = (blockIdx.x * blockDim.x + threadIdx.x) >> 5;   // uniform per wave
    int lane = threadIdx.x & 31;
    if (wave >= N_NODES / 16) return;                          // whole-wave exit

    int m    = lane & 15;        // matrix row this lane feeds
    int half = lane >> 4;        // 0: K=0,1   1: K=2,3
    int node = wave * 16 + m;
    const float* hrow = h + (size_t)node * D_FEAT;

    v2f ones = {1.0f, 1.0f};
    v8f accD = {0.f,0.f,0.f,0.f,0.f,0.f,0.f,0.f};
    v8f accS = accD;

    for (int kb = 0; kb < D_FEAT; kb += 4) {
        int k = kb + 2 * half;
        float2 hv = *(const float2*)(hrow + k);
        float2 wd = *(const float2*)(gw + k);            // w_dst slice
        float2 ws = *(const float2*)(gw + D_FEAT + k);   // w_src slice
        v2f aD = { hv.x * wd.x, hv.y * wd.y };
        v2f aS = { hv.x * ws.x, hv.y * ws.y };
        accD = __builtin_amdgcn_wmma_f32_16x16x4_f32(
            false, aD, false, ones, (short)0, accD, false, false);
        accS = __builtin_amdgcn_wmma_f32_16x16x4_f32(
            false, aS, false, ones, (short)0, accS, false, false);
    }

    // Extract column N=0: lane 0 holds rows M=0..7, lane 16 holds M=8..15.
    if (m == 0) {
        float b = gb[0];
        int base = wave * 16 + 8 * half;
        #pragma unroll
        for (int r = 0; r < 8; ++r) {
            sd[base + r] = accD[r] + b;
            ss[base + r] = accS[r];
        }
    }
}

// ---------------------------------------------------------------------------
// Kernel 3: histogram of dst.
// ---------------------------------------------------------------------------
__global__ void k_hist(const long long* __restrict__ dst,
                       unsigned int* __restrict__ cnt) {
    int i = blockIdx.x * blockDim.x + threadIdx.x;
    if (i < N_EDGES) atomicAdd(&cnt[(int)dst[i]], 1u);
}

// ---------------------------------------------------------------------------
// Kernel 4: single-block exclusive scan over 10000 counts -> offsets/cursor.
// 1024 threads x 10 buckets each; Hillis-Steele over per-thread partials.
// ---------------------------------------------------------------------------
__global__ __launch_bounds__(1024)
void k_scan(const unsigned int* __restrict__ cnt,
            int* __restrict__ offs,
            unsigned int* __restrict__ cur) {
    __shared__ unsigned int s[1024];
    const int C = 10;                       // 1024*10 = 10240 >= N_NODES
    int t = threadIdx.x;
    int base = t * C;

    unsigned int part = 0;
    for (int i = 0; i < C; ++i) {
        int idx = base + i;
        if (idx < N_NODES) part += cnt[idx];
    }
    s[t] = part;
    __syncthreads();
    for (int off = 1; off < 1024; off <<= 1) {
        unsigned int v = (t >= off) ? s[t - off] : 0u;
        __syncthreads();
        s[t] += v;
        __syncthreads();
    }
    unsigned int run = s[t] - part;         // exclusive prefix for this thread
    for (int i = 0; i < C; ++i) {
        int idx = base + i;
        if (idx < N_NODES) {
            offs[idx] = (int)run;
            cur[idx]  = run;
            run += cnt[idx];
        }
    }
    if (t == 1023) offs[N_NODES] = (int)s[1023];   // total = N_EDGES
}

// ---------------------------------------------------------------------------
// Kernel 5: compute per-edge gate e and scatter (e, src) into CSR order.
// ---------------------------------------------------------------------------
__global__ void k_scatter(const long long* __restrict__ src,
                          const long long* __restrict__ dst,
                          const float* __restrict__ sd,
                          const float* __restrict__ ss,
                          const float* __restrict__ dn,
                          unsigned int* __restrict__ cur,
                          float* __restrict__ eval,
                          int* __restrict__ esrc) {
    int i = blockIdx.x * blockDim.x + threadIdx.x;
    if (i >= N_EDGES) return;
    int s = (int)src[i];
    int d = (int)dst[i];
    float g = tanhf(sd[d] + ss[s]);
    float e = g * dn[d] * dn[s];
    unsigned int p = atomicAdd(&cur[d], 1u);
    eval[p] = e;
    esrc[p] = s;
}

// ---------------------------------------------------------------------------
// Kernel 6: one wave per dst node; register accumulation, zero output atomics.
// Lane owns 8 contiguous features. Edge metadata read 32-at-a-time and
// broadcast across the wave32 with __shfl. h rows served from L2 (10 MB).
// ---------------------------------------------------------------------------
__global__ void k_gather(const float* __restrict__ h,
                         const int* __restrict__ offs,
                         const float* __restrict__ eval,
                         const int* __restrict__ esrc,
                         float* __restrict__ z) {
    int wave = (blockIdx.x * blockDim.x + threadIdx.x) >> 5;
    int lane = threadIdx.x & 31;
    if (wave >= N_NODES) return;

    int beg = offs[wave];
    int end = offs[wave + 1];

    float4 a0 = {0.f, 0.f, 0.f, 0.f};
    float4 a1 = {0.f, 0.f, 0.f, 0.f};
    const float* hb = h + lane * 8;

    for (int i = beg; i < end; i += 32) {
        int   idx = i + lane;
        int   ms  = (idx < end) ? esrc[idx] : 0;
        float me  = (idx < end) ? eval[idx] : 0.0f;
        int nval  = end - i; if (nval > 32) nval = 32;
        for (int j = 0; j < nval; ++j) {
            int   sn = __shfl(ms, j, 32);
            float ev = __shfl(me, j, 32);
            const float4* hp = (const float4*)(hb + (size_t)sn * D_FEAT);
            float4 h0 = hp[0];
            float4 h1 = hp[1];
            a0.x = fmaf(ev, h0.x, a0.x);
            a0.y = fmaf(ev, h0.y, a0.y);
            a0.z = fmaf(ev, h0.z, a0.z);
            a0.w = fmaf(ev, h0.w, a0.w);
            a1.x = fmaf(ev, h1.x, a1.x);
            a1.y = fmaf(ev, h1.y, a1.y);
            a1.z = fmaf(ev, h1.z, a1.z);
            a1.w = fmaf(ev, h1.w, a1.w);
        }
    }

    float4* zp = (float4*)(z + (size_t)wave * D_FEAT + lane * 8);
    zp[0] = a0;
    zp[1] = a1;
}

// ---------------------------------------------------------------------------
// Launcher
// ---------------------------------------------------------------------------
extern "C" void kernel_launch(void* const* d_in, const int* in_sizes, int n_in,
                              void* d_out, int out_size, void* d_ws, size_t ws_size,
                              hipStream_t stream) {
    const float*     h      = (const float*)d_in[0];       // [10000, 256]
    const float*     d_norm = (const float*)d_in[1];       // [10000]
    const float*     gate_w = (const float*)d_in[2];       // [1, 512]
    const float*     gate_b = (const float*)d_in[3];       // [1]
    const long long* src    = (const long long*)d_in[4];   // [320000] int64
    const long long* dst    = (const long long*)d_in[5];   // [320000] int64
    float*           z      = (float*)d_out;               // [10000, 256]

    // Workspace layout (bytes); total ~2.77 MB.
    char* ws = (char*)d_ws;
    float*        sd   = (float*)(ws + 0);          // 10000 floats
    float*        ss   = (float*)(ws + 40960);      // 10000 floats
    unsigned int* cnt  = (unsigned int*)(ws + 81920);   // 10000 u32
    int*          offs = (int*)(ws + 122880);       // 10001 ints
    unsigned int* cur  = (unsigned int*)(ws + 163840);  // 10000 u32
    float*        eval = (float*)(ws + 204800);     // 320000 floats
    int*          esrc = (int*)(ws + 1484800);      // 320000 ints

    // 1) zero histogram
    k_zero<<<(N_NODES + 255) / 256, 256, 0, stream>>>(cnt);

    // 2) node scores (WMMA): 625 waves of 16 nodes
    k_scores<<<(625 * 32 + 255) / 256, 256, 0, stream>>>(h, gate_w, gate_b, sd, ss);

    // 3) dst histogram
    k_hist<<<(N_EDGES + 255) / 256, 256, 0, stream>>>(dst, cnt);

    // 4) exclusive scan -> row offsets + scatter cursors
    k_scan<<<1, 1024, 0, stream>>>(cnt, offs, cur);

    // 5) per-edge gate + CSR scatter
    k_scatter<<<(N_EDGES + 255) / 256, 256, 0, stream>>>(src, dst, sd, ss, d_norm,
                                                         cur, eval, esrc);

    // 6) per-dst register accumulation (no output atomics)
    k_gather<<<(N_NODES * 32 + 255) / 256, 256, 0, stream>>>(h, offs, eval, esrc, z);
}